// RoiAlign_37529424232843
// MI455X (gfx1250) — compile-verified
//
#include <hip/hip_runtime.h>
#include <math.h>
#include <stdint.h>

#define CROP 14
#define NPIX (CROP * CROP)   // 196
#define CCH 256              // channels
#define THREADS 256          // 8 wave32s
#define PIX_PER_ITER 4       // 4 pixels in flight, 64 lanes (float4) per pixel
#define NITER (NPIX / PIX_PER_ITER)  // 49

// ---- CDNA5 async copy path (guarded: falls back to direct loads + prefetch) ----
#if defined(__AMDGCN__) && __has_builtin(__builtin_amdgcn_global_load_async_to_lds_b128) && \
    __has_builtin(__builtin_amdgcn_s_wait_asynccnt)
#define ROI_ASYNC 1
#else
#define ROI_ASYNC 0
#endif

#if ROI_ASYNC
// Param 0 of the builtin is 'int __vector(4) __device__ *' (addrspace(1)),
// param 1 is the LDS-side pointer (addrspace(3)); offset/cpol are immediates.
typedef int roi_v4i __attribute__((ext_vector_type(4)));
typedef __attribute__((address_space(1))) roi_v4i* roi_gptr;
typedef __attribute__((address_space(3))) roi_v4i* roi_lptr;

__device__ __forceinline__ void roi_async_cp16(const float* g, float* l) {
  // global_load_async_to_lds_b128: 16B per lane, tracked by ASYNCcnt.
  // Generic LDS pointer low 32 bits == LDS byte offset (ISA §10.2 aperture rules).
  // cpol=0 (RT): feature maps (~179 MB total) should stay resident in the 192 MB L2.
  __builtin_amdgcn_global_load_async_to_lds_b128(
      (roi_gptr)(uintptr_t)g,
      (roi_lptr)(uint32_t)(uintptr_t)l,
      /*offset=*/0, /*cpol=*/0);
}
#endif

// Native clang vector (HIP's float4 is a class and is rejected by the builtin).
typedef float roi_v4f __attribute__((ext_vector_type(4)));

__device__ __forceinline__ void roi_store_nt(float* p, float4 v) {
#if defined(__AMDGCN__) && __has_builtin(__builtin_nontemporal_store)
  // Output (205 MB, write-once) is streamed with NT hint so it does not evict
  // the FPN feature maps from L2.
  roi_v4f nv;
  nv.x = v.x; nv.y = v.y; nv.z = v.z; nv.w = v.w;
  __builtin_nontemporal_store(nv, (roi_v4f*)p);
#else
  *(float4*)p = v;
#endif
}

__global__ __launch_bounds__(THREADS) void RoiAlignFpn_kernel(
    const int* __restrict__ image_shape,
    const float* __restrict__ boxes,
    const float* __restrict__ fpn0,
    const float* __restrict__ fpn1,
    const float* __restrict__ fpn2,
    const float* __restrict__ fpn3,
    const float* __restrict__ fpn4,
    float* __restrict__ out,
    int n_per_batch) {
  const int box_id = blockIdx.x;
  const int b = box_id / n_per_batch;

  const float bx1 = boxes[box_id * 4 + 0];
  const float by1 = boxes[box_id * 4 + 1];
  const float bx2 = boxes[box_id * 4 + 2];
  const float by2 = boxes[box_id * 4 + 3];
  const float img_h = (float)image_shape[1];
  const float img_w = (float)image_shape[2];

  // levels = clip(floor(1 + log(size/224 + 1e-7)/log(2)), 0, 4)
  const float bsize = sqrtf((bx2 - bx1) * (by2 - by1));
  float lvl = floorf(1.0f + logf(bsize / 224.0f + 1e-7f) * 1.44269504088896340736f);
  lvl = fminf(fmaxf(lvl, 0.0f), 4.0f);
  const int li = (int)lvl;

  int S;
  const float* f;
  switch (li) {
    case 0:  S = 256; f = fpn0; break;
    case 1:  S = 128; f = fpn1; break;
    case 2:  S = 64;  f = fpn2; break;
    case 3:  S = 32;  f = fpn3; break;
    default: S = 16;  f = fpn4; break;
  }
  const float* __restrict__ fb = f + (size_t)b * (size_t)S * (size_t)S * CCH;
  const float Hf = (float)S;  // square features: Hf == Wf

  __shared__ int sy0[CROP], sy1[CROP], sx0[CROP], sx1[CROP];
  __shared__ float swy[CROP], swx[CROP];
  __shared__ unsigned char syv[CROP], sxv[CROP];

  const int tid = threadIdx.x;
  if (tid < CROP) {
    // normalized box (y1n, x1n, y2n, x2n) per reference
    const float y1n = by1 / img_h * Hf / (Hf - 1.0f);
    const float x1n = bx1 / img_w * Hf / (Hf - 1.0f);
    const float y2n = (by2 / img_h * Hf - 1.0f) / (Hf - 1.0f);
    const float x2n = (bx2 / img_w * Hf - 1.0f) / (Hf - 1.0f);
    const float ys = y1n * (Hf - 1.0f) +
                     (float)tid * ((y2n - y1n) * (Hf - 1.0f) / (float)(CROP - 1));
    const float xs = x1n * (Hf - 1.0f) +
                     (float)tid * ((x2n - x1n) * (Hf - 1.0f) / (float)(CROP - 1));
    const float y0 = floorf(ys);
    const float x0 = floorf(xs);
    swy[tid] = ys - y0;
    swx[tid] = xs - x0;
    sy0[tid] = (int)fminf(fmaxf(y0, 0.0f), Hf - 1.0f);
    sy1[tid] = (int)fminf(fmaxf(y0 + 1.0f, 0.0f), Hf - 1.0f);
    sx0[tid] = (int)fminf(fmaxf(x0, 0.0f), Hf - 1.0f);
    sx1[tid] = (int)fminf(fmaxf(x0 + 1.0f, 0.0f), Hf - 1.0f);
    syv[tid] = (ys >= 0.0f) && (ys <= Hf - 1.0f);
    sxv[tid] = (xs >= 0.0f) && (xs <= Hf - 1.0f);
  }
  __syncthreads();

  const int psub = tid >> 6;        // pixel slot within group of 4
  const int c = (tid & 63) * 4;     // channel offset (float4 chunk)
  float* __restrict__ outb = out + (size_t)box_id * NPIX * CCH;

  auto corner = [&](int y, int x) -> const float* {
    return fb + (size_t)(y * S + x) * CCH + c;
  };

  auto emit = [&](int p, int j, int k, float4 tl, float4 tr, float4 bl, float4 br) {
    const float wx = swx[k];
    const float wy = swy[j];
    float4 o;
    {
      float t0 = tl.x + (tr.x - tl.x) * wx;
      float b0 = bl.x + (br.x - bl.x) * wx;
      o.x = t0 + (b0 - t0) * wy;
      t0 = tl.y + (tr.y - tl.y) * wx;
      b0 = bl.y + (br.y - bl.y) * wx;
      o.y = t0 + (b0 - t0) * wy;
      t0 = tl.z + (tr.z - tl.z) * wx;
      b0 = bl.z + (br.z - bl.z) * wx;
      o.z = t0 + (b0 - t0) * wy;
      t0 = tl.w + (tr.w - tl.w) * wx;
      b0 = bl.w + (br.w - bl.w) * wx;
      o.w = t0 + (b0 - t0) * wy;
    }
    if (!(syv[j] & sxv[k])) { o.x = 0.0f; o.y = 0.0f; o.z = 0.0f; o.w = 0.0f; }
    roi_store_nt(outb + (size_t)p * CCH + c, o);
  };

#if ROI_ASYNC
  // Double-buffered async staging: 2 x (4 pixels x 4 corners x 256 f32) = 32 KB LDS.
  __shared__ __align__(16) float abuf[2][PIX_PER_ITER][4][CCH];

  auto stage = [&](int buf, int p) {
    const int j = p / CROP, k = p % CROP;
    float* lb = &abuf[buf][psub][0][c];
    roi_async_cp16(corner(sy0[j], sx0[k]), lb + 0 * CCH);
    roi_async_cp16(corner(sy0[j], sx1[k]), lb + 1 * CCH);
    roi_async_cp16(corner(sy1[j], sx0[k]), lb + 2 * CCH);
    roi_async_cp16(corner(sy1[j], sx1[k]), lb + 3 * CCH);
  };

  int p = psub;
  int buf = 0;
  stage(0, p);
#pragma unroll 1
  for (int it = 0; it < NITER; ++it) {
    const int pn = p + PIX_PER_ITER;
    if (it + 1 < NITER) {
      stage(buf ^ 1, pn);                      // prefetch next group (ASYNCcnt += 4)
      __builtin_amdgcn_s_wait_asynccnt(4);     // wait: current group's 4 copies done
    } else {
      __builtin_amdgcn_s_wait_asynccnt(0);
    }
    {
      const int j = p / CROP, k = p % CROP;
      const float* lb = &abuf[buf][psub][0][c];
      const float4 tl = *(const float4*)(lb + 0 * CCH);
      const float4 tr = *(const float4*)(lb + 1 * CCH);
      const float4 bl = *(const float4*)(lb + 2 * CCH);
      const float4 br = *(const float4*)(lb + 3 * CCH);
      emit(p, j, k, tl, tr, bl, br);
    }
    p = pn;
    buf ^= 1;
  }
#else
  // Fallback: direct coalesced float4 gathers + gfx1250 global_prefetch of next group.
#pragma unroll 1
  for (int p = psub; p < NPIX; p += PIX_PER_ITER) {
    const int j = p / CROP, k = p % CROP;
    const float4 tl = *(const float4*)corner(sy0[j], sx0[k]);
    const float4 tr = *(const float4*)corner(sy0[j], sx1[k]);
    const float4 bl = *(const float4*)corner(sy1[j], sx0[k]);
    const float4 br = *(const float4*)corner(sy1[j], sx1[k]);
    const int pn = p + PIX_PER_ITER;
    if (pn < NPIX) {
      const int j2 = pn / CROP, k2 = pn % CROP;
      __builtin_prefetch(corner(sy0[j2], sx0[k2]), 0, 3);
      __builtin_prefetch(corner(sy0[j2], sx1[k2]), 0, 3);
      __builtin_prefetch(corner(sy1[j2], sx0[k2]), 0, 3);
      __builtin_prefetch(corner(sy1[j2], sx1[k2]), 0, 3);
    }
    emit(p, j, k, tl, tr, bl, br);
  }
#endif
}

extern "C" void kernel_launch(void* const* d_in, const int* in_sizes, int n_in,
                              void* d_out, int out_size, void* d_ws, size_t ws_size,
                              hipStream_t stream) {
  (void)n_in; (void)out_size; (void)d_ws; (void)ws_size;
  const int* image_shape = (const int*)d_in[0];
  const float* boxes = (const float*)d_in[1];
  // d_in[2] (scores) is unused by the reference output.
  const float* f0 = (const float*)d_in[3];
  const float* f1 = (const float*)d_in[4];
  const float* f2 = (const float*)d_in[5];
  const float* f3 = (const float*)d_in[6];
  const float* f4 = (const float*)d_in[7];
  float* out = (float*)d_out;

  const int total_boxes = in_sizes[1] / 4;                 // B*N
  int Bc = in_sizes[3] / (256 * 256 * 256);                // fpn0 = B*256*256*C
  if (Bc < 1) Bc = 1;
  const int n_per_batch = total_boxes / Bc;

  RoiAlignFpn_kernel<<<dim3(total_boxes), dim3(THREADS), 0, stream>>>(
      image_shape, boxes, f0, f1, f2, f3, f4, out, n_per_batch);
}